// CCFG_8504035246177
// MI455X (gfx1250) — compile-verified
//
#include <hip/hip_runtime.h>
#include <hip/hip_bf16.h>

// Problem constants
#define NB 32      // batch N
#define CC 256     // channels C
#define TT 128     // time T
#define VV 49      // vertices V
#define RR 32      // rel channels R = C/8
#define OO 256     // out channels O = C
#define NCLS 12
#define BN_EPS 1e-5f

typedef float v2f __attribute__((ext_vector_type(2)));
typedef float v8f __attribute__((ext_vector_type(8)));

// ---------------------------------------------------------------------------
// Kernel 1: temporal mean  x(N,C,T,V) -> x_tm(N,C,V).  The ONLY pass over the
// 205 MB input; everything else is tiny.  One block per (n,c) slab of
// T*V = 6272 contiguous floats.  Active lanes 0..195 read address
// base + tid*4 each iteration -> fully contiguous wave loads.
// ---------------------------------------------------------------------------
__global__ void k1_mean_t(const float* __restrict__ x, float* __restrict__ xtm) {
    __shared__ float part[4 * VV];
    const int nc  = blockIdx.x;                 // 0 .. N*C-1
    const int tid = threadIdx.x;                // blockDim = 224 (7 waves)
    const float* base = x + (size_t)nc * (TT * VV);
    if (tid < 4 * VV) {
        float acc = 0.f;
        const float* p = base + tid;            // covers 4 t-rows x 49 v
        #pragma unroll 8
        for (int j = 0; j < TT / 4; ++j) acc += p[j * 4 * VV];
        part[tid] = acc;
    }
    __syncthreads();
    if (tid < VV) {
        float s = part[tid] + part[tid + VV] + part[tid + 2 * VV] + part[tid + 3 * VV];
        xtm[(size_t)nc * VV + tid] = s * (1.0f / TT);
    }
}

// ---------------------------------------------------------------------------
// Kernel 2: per batch n -> x1[r,u], x2[r,v] (R x V small GEMVs) then
// relsum[n,r,v] = sum_u tanh(x1[r,u] - x2[r,v]).  One block per n.
// ---------------------------------------------------------------------------
__global__ void k2_relsum(const float* __restrict__ xtm,
                          const float* __restrict__ w1, const float* __restrict__ b1,
                          const float* __restrict__ w2, const float* __restrict__ b2,
                          float* __restrict__ relsum) {
    __shared__ float x12[2 * RR * VV];          // 3136 floats = 12.25 KB
    const int n = blockIdx.x;
    const float* xb = xtm + (size_t)n * CC * VV;
    for (int i = threadIdx.x; i < 2 * RR * VV; i += blockDim.x) {
        const int sel = i / (RR * VV);
        const int rem = i - sel * (RR * VV);
        const int r = rem / VV;
        const int u = rem - r * VV;
        const float* w = sel ? w2 : w1;
        float s = sel ? b2[r] : b1[r];
        const float* wr = w + (size_t)r * CC;
        for (int c = 0; c < CC; ++c) s += wr[c] * xb[(size_t)c * VV + u];
        x12[i] = s;
    }
    __syncthreads();
    const float* x1 = x12;
    const float* x2 = x12 + RR * VV;
    for (int i = threadIdx.x; i < RR * VV; i += blockDim.x) {
        const int r = i / VV;
        const int v = i - r * VV;
        const float rv = x2[r * VV + v];
        float s = 0.f;
        for (int u = 0; u < VV; ++u) s += tanhf(x1[r * VV + u] - rv);
        relsum[(size_t)n * RR * VV + i] = s;
    }
}

// ---------------------------------------------------------------------------
// Kernel 3: x3m[n,o,v] = sum_c w3[o,c]*xtm[n,c,v] + b3[o]
// FP32 WMMA 16x16x4.  One wave per (n, o-tile, v-tile); v padded 49->64 with
// branch-free masked loads so EXEC is all-1s at every v_wmma.
//   A (16x4 f32): lane l -> M=l&15, K = k0 + 2*(l>>4) + {0,1}
//   B (4x16 f32): lane l -> N=l&15, same K pair
//   C/D:          lane l -> N=l&15, VGPR r -> M = r + 8*(l>>4)
// ---------------------------------------------------------------------------
__global__ void k3_x3m_wmma(const float* __restrict__ xtm,
                            const float* __restrict__ w3, const float* __restrict__ b3,
                            float* __restrict__ x3m) {
    const int vt   = blockIdx.x;                // v tile 0..3
    const int mt   = blockIdx.y;                // o tile 0..15
    const int nb   = blockIdx.z;                // batch
    const int lane = threadIdx.x;               // 0..31
    const int half = lane >> 4;
    const int l15  = lane & 15;
    const int orow = mt * 16 + l15;
    const int v    = vt * 16 + l15;
    const float mask = (v < VV) ? 1.f : 0.f;
    const int vc = (v < VV) ? v : 0;            // clamped, branch-free

    const float* arow = w3 + (size_t)orow * CC;               // A row (contig in k)
    const float* bcol = xtm + (size_t)nb * CC * VV + vc;      // B col (stride V in k)

    v8f acc = {};
    for (int k0 = 0; k0 < CC; k0 += 4) {
        const int k = k0 + 2 * half;
        v2f a; a.x = arow[k];                 a.y = arow[k + 1];
        v2f b; b.x = bcol[(size_t)k * VV] * mask;
               b.y = bcol[(size_t)(k + 1) * VV] * mask;
        acc = __builtin_amdgcn_wmma_f32_16x16x4_f32(
                  false, a, false, b, (short)0, acc, false, false);
    }
    if (v < VV) {
        #pragma unroll
        for (int r = 0; r < 8; ++r) {
            const int o = mt * 16 + r + 8 * half;
            x3m[((size_t)nb * OO + o) * VV + v] = acc[r] + b3[o];
        }
    }
}

// ---------------------------------------------------------------------------
// Kernel 4: pooled[n,o] = (1/V) * sum_v x3m[n,o,v] *
//                          ( V*b4[o] + sum_r w4[o,r]*relsum[n,r,v] )
// One block per n, thread = o.
// ---------------------------------------------------------------------------
__global__ void k4_pooled(const float* __restrict__ relsum,
                          const float* __restrict__ w4, const float* __restrict__ b4,
                          const float* __restrict__ x3m, float* __restrict__ pooled) {
    __shared__ float rl[RR * VV];               // 1568 floats
    const int n = blockIdx.x;
    const int o = threadIdx.x;                  // 256 threads
    for (int i = threadIdx.x; i < RR * VV; i += blockDim.x)
        rl[i] = relsum[(size_t)n * RR * VV + i];
    __syncthreads();

    float w4l[RR];
    #pragma unroll
    for (int r = 0; r < RR; ++r) w4l[r] = w4[(size_t)o * RR + r];
    const float vb4 = (float)VV * b4[o];

    const float* x3 = x3m + ((size_t)n * OO + o) * VV;
    float acc = 0.f;
    for (int v = 0; v < VV; ++v) {
        float s = vb4;
        #pragma unroll
        for (int r = 0; r < RR; ++r) s += w4l[r] * rl[r * VV + v];
        acc += x3[v] * s;
    }
    pooled[(size_t)n * OO + o] = acc * (1.0f / VV);
}

// ---------------------------------------------------------------------------
// Kernel 5: h(32x256) = pooled(32x256) @ fc_w^T(256x256) + fc_b, FP32 WMMA.
// B[k][j] = fc_w[j][k] -> per-lane contiguous k loads.  Grid (16 jt, 2 mt).
// ---------------------------------------------------------------------------
__global__ void k5_fc_wmma(const float* __restrict__ pooled,
                           const float* __restrict__ fc_w, const float* __restrict__ fc_b,
                           float* __restrict__ h) {
    const int jt   = blockIdx.x;                // 0..15
    const int mt   = blockIdx.y;                // 0..1
    const int lane = threadIdx.x;
    const int half = lane >> 4;
    const int l15  = lane & 15;
    const int nrow = mt * 16 + l15;
    const int j    = jt * 16 + l15;

    const float* arow = pooled + (size_t)nrow * CC;
    const float* brow = fc_w + (size_t)j * CC;

    v8f acc = {};
    for (int k0 = 0; k0 < CC; k0 += 4) {
        const int k = k0 + 2 * half;
        v2f a; a.x = arow[k]; a.y = arow[k + 1];
        v2f b; b.x = brow[k]; b.y = brow[k + 1];
        acc = __builtin_amdgcn_wmma_f32_16x16x4_f32(
                  false, a, false, b, (short)0, acc, false, false);
    }
    #pragma unroll
    for (int r = 0; r < 8; ++r) {
        const int nn = mt * 16 + r + 8 * half;
        h[(size_t)nn * CC + j] = acc[r] + fc_b[j];
    }
}

// ---------------------------------------------------------------------------
// Kernel 6: BatchNorm over batch dim (biased var, N=32 == wave32!) + ReLU +
// broadcast to 12 classes.  Block per channel j, lane = n, wave reductions.
// ---------------------------------------------------------------------------
__global__ void k6_bn_out(const float* __restrict__ h,
                          const float* __restrict__ g, const float* __restrict__ b,
                          float* __restrict__ out) {
    const int j = blockIdx.x;                   // 0..255
    const int n = threadIdx.x;                  // 0..31 (one wave)
    const float v = h[(size_t)n * CC + j];
    float s = v, sq = v * v;
    #pragma unroll
    for (int off = 16; off > 0; off >>= 1) {
        s  += __shfl_xor(s,  off, 32);
        sq += __shfl_xor(sq, off, 32);
    }
    const float mu  = s * (1.0f / NB);
    const float var = sq * (1.0f / NB) - mu * mu;
    float y = (v - mu) * rsqrtf(var + BN_EPS) * g[j] + b[j];
    y = fmaxf(y, 0.f);
    #pragma unroll
    for (int cls = 0; cls < NCLS; ++cls)
        out[((size_t)n * NCLS + cls) * CC + j] = y;
}

// ---------------------------------------------------------------------------
extern "C" void kernel_launch(void* const* d_in, const int* in_sizes, int n_in,
                              void* d_out, int out_size, void* d_ws, size_t ws_size,
                              hipStream_t stream) {
    const float* x    = (const float*)d_in[0];
    const float* w1   = (const float*)d_in[1];
    const float* b1   = (const float*)d_in[2];
    const float* w2   = (const float*)d_in[3];
    const float* b2   = (const float*)d_in[4];
    const float* w3   = (const float*)d_in[5];
    const float* b3   = (const float*)d_in[6];
    const float* w4   = (const float*)d_in[7];
    const float* b4   = (const float*)d_in[8];
    const float* fcw  = (const float*)d_in[9];
    const float* fcb  = (const float*)d_in[10];
    const float* bng  = (const float*)d_in[11];
    const float* bnb  = (const float*)d_in[12];
    float* out = (float*)d_out;

    // workspace layout (floats)
    float* ws     = (float*)d_ws;
    float* xtm    = ws;                                  // N*C*V  = 401408
    float* relsum = xtm    + (size_t)NB * CC * VV;       // N*R*V  =  50176
    float* x3m    = relsum + (size_t)NB * RR * VV;       // N*O*V  = 401408
    float* pooled = x3m    + (size_t)NB * OO * VV;       // N*O    =   8192
    float* h      = pooled + (size_t)NB * OO;            // N*C    =   8192

    // 1. temporal mean (bandwidth-dominant pass)
    k1_mean_t<<<NB * CC, 224, 0, stream>>>(x, xtm);
    // 2. x1/x2 + relation column sums
    k2_relsum<<<NB, 256, 0, stream>>>(xtm, w1, b1, w2, b2, relsum);
    // 3. x3m GEMM via fp32 WMMA
    k3_x3m_wmma<<<dim3(4, 16, NB), 32, 0, stream>>>(xtm, w3, b3, x3m);
    // 4. pooled contraction
    k4_pooled<<<NB, 256, 0, stream>>>(relsum, w4, b4, x3m, pooled);
    // 5. fc GEMM via fp32 WMMA
    k5_fc_wmma<<<dim3(16, 2), 32, 0, stream>>>(pooled, fcw, fcb, h);
    // 6. batchnorm + relu + class broadcast
    k6_bn_out<<<CC, 32, 0, stream>>>(h, bng, bnb, out);
}